// GATBase_3848290697849
// MI455X (gfx1250) — compile-verified
//
#include <hip/hip_runtime.h>
#include <hip/hip_bf16.h>
#include <math.h>

#define NNODES 10000
#define INCH   128
#define HC     128      // HEADS*HID == conv2 width == 128
#define HEADS1 4
#define HID1   32
#define OUTCH  40
#define NEGS   0.2f

typedef __attribute__((ext_vector_type(2))) float v2f;
typedef __attribute__((ext_vector_type(8))) float v8f;

// ---------------------------------------------------------------- utilities

__global__ void fill_kernel(float* __restrict__ p, float v, int count) {
    int i = blockIdx.x * blockDim.x + threadIdx.x;
    if (i < count) p[i] = v;
}

__device__ __forceinline__ void atomicMaxF32(float* addr, float val) {
    // sign-split trick: works when slot is initialized to -inf
    if (val >= 0.0f) {
        atomicMax((int*)addr, __float_as_int(val));
    } else {
        atomicMin((unsigned int*)addr, (unsigned int)__float_as_int(val));
    }
}

// ------------------------------------------------- WMMA GEMM: H = X @ W
// X: [nrows x 128] row-major, W: [128 x 128] row-major, H: [nrows x 128].
// Each wave computes one 16-row x 128-col strip: 8 accumulators, 32 k-steps
// of K=4 -> 256 v_wmma_f32_16x16x4_f32 per wave.
//
// W is staged in LDS in WMMA-native pair layout:
//   Wlds[p][n'] = {W[2p][n], W[2p+1][n]},  n' = (n + (p&1)*16) & 127
// so the B operand is ONE ds_load_b64 straight into an even VGPR pair
// (no packing moves), and since p&1 == lane-half for every k-step, the
// +16-column swizzle puts lanes 0-15 on banks 0-31 and lanes 16-31 on
// banks 32-63: conflict-free on the 64-bank LDS.
__global__ __launch_bounds__(128) void gemm128_wmma(const float* __restrict__ X,
                                                    const float* __restrict__ W,
                                                    float* __restrict__ H,
                                                    int nrows) {
    __shared__ float2 Wlds[64 * 128];   // 64 KB
    int t = threadIdx.x;

    // staging: 64 k-pairs x 32 col-groups-of-4 = 2048 units, 16 per thread
    #pragma unroll
    for (int u = 0; u < 16; ++u) {
        int i = t + u * 128;
        int p = i >> 5;                 // k pair index (0..63)
        int g = (i & 31) << 2;          // column group base (0,4,...,124)
        float4 r0 = *(const float4*)(W + (long)(2 * p)     * 128 + g);
        float4 r1 = *(const float4*)(W + (long)(2 * p + 1) * 128 + g);
        int gsw = (g + (p & 1) * 16) & 127;   // swizzled column base
        float2* dp = &Wlds[p * 128 + gsw];
        dp[0] = make_float2(r0.x, r1.x);
        dp[1] = make_float2(r0.y, r1.y);
        dp[2] = make_float2(r0.z, r1.z);
        dp[3] = make_float2(r0.w, r1.w);
    }
    __syncthreads();

    int wave = t >> 5;
    int lane = t & 31;
    int strip = blockIdx.x * 4 + wave;            // 16-row output strip
    if (strip * 16 >= nrows) return;

    int mrow = strip * 16 + (lane & 15);          // A-matrix row for this lane
    int half = lane >> 4;                         // K sub-pair selector
    int ncol = lane & 15;                         // B/C column within tile

    v8f acc[8];
    #pragma unroll
    for (int nt = 0; nt < 8; ++nt) acc[nt] = (v8f){0,0,0,0,0,0,0,0};

    const float* xrow = X + (long)mrow * 128;
    #pragma unroll 4
    for (int k0 = 0; k0 < 128; k0 += 4) {
        int kk = k0 + 2 * half;                   // even; pair p = kk>>1, p&1==half
        float2 av = *(const float2*)(xrow + kk);  // A: lane {K=kk, K=kk+1}
        v2f a; a.x = av.x; a.y = av.y;
        const float2* brow = &Wlds[(kk >> 1) * 128];
        #pragma unroll
        for (int nt = 0; nt < 8; ++nt) {
            // swizzled column for tile nt: ((nt+half)&7)*16 + ncol
            float2 bv = brow[(((nt + half) & 7) << 4) + ncol];
            v2f b; b.x = bv.x; b.y = bv.y;
            acc[nt] = __builtin_amdgcn_wmma_f32_16x16x4_f32(
                false, a, false, b, (short)0, acc[nt], false, false);
        }
    }

    // C/D layout: VGPR i -> row i + 8*half, column = lane&15
    #pragma unroll
    for (int nt = 0; nt < 8; ++nt) {
        float* op = H + (long)(strip * 16 + 8 * half) * 128 + nt * 16 + ncol;
        #pragma unroll
        for (int i = 0; i < 8; ++i)
            op[(long)i * 128] = acc[nt][i];
    }
}

// ------------------------------------- per-node attention coefficients
// asrc[n,h] = dot(hfeat[n,h,:], a_src[h,:]);  adst likewise.
__global__ void att_coeff_kernel(const float* __restrict__ hfeat,
                                 const float* __restrict__ a_src,
                                 const float* __restrict__ a_dst,
                                 float* __restrict__ asrc,
                                 float* __restrict__ adst,
                                 int n, int heads, int d) {
    int idx = blockIdx.x * blockDim.x + threadIdx.x;
    if (idx >= n * heads) return;
    int node = idx / heads, hd = idx % heads;
    const float* hp = hfeat + (long)node * HC + hd * d;
    const float* as = a_src + hd * d;
    const float* ad = a_dst + hd * d;
    float s = 0.f, tsum = 0.f;
    for (int i = 0; i < d; i += 4) {
        float4 hv = *(const float4*)(hp + i);
        float4 sv = *(const float4*)(as + i);
        float4 dv = *(const float4*)(ad + i);
        s    += hv.x*sv.x + hv.y*sv.y + hv.z*sv.z + hv.w*sv.w;
        tsum += hv.x*dv.x + hv.y*dv.y + hv.z*dv.z + hv.w*dv.w;
    }
    asrc[idx] = s;
    adst[idx] = tsum;
}

// ------------------------------------------------- edge pass 1: segment max
__global__ void edge_max_kernel(const int* __restrict__ src,
                                const int* __restrict__ dst,
                                const float* __restrict__ asrc,
                                const float* __restrict__ adst,
                                float* __restrict__ mbuf,
                                int E, int n, int heads) {
    int e = blockIdx.x * blockDim.x + threadIdx.x;
    if (e >= E + n) return;
    int s  = (e < E) ? src[e] : (e - E);     // self-loops appended
    int dn = (e < E) ? dst[e] : (e - E);
    for (int h = 0; h < heads; ++h) {
        float v = asrc[s * heads + h] + adst[dn * heads + h];
        v = (v > 0.f) ? v : NEGS * v;
        atomicMaxF32(&mbuf[dn * heads + h], v);
    }
}

// ------------------------------------------------- edge pass 2: segment sum
__global__ void edge_sum_kernel(const int* __restrict__ src,
                                const int* __restrict__ dst,
                                const float* __restrict__ asrc,
                                const float* __restrict__ adst,
                                const float* __restrict__ mbuf,
                                float* __restrict__ zbuf,
                                int E, int n, int heads) {
    int e = blockIdx.x * blockDim.x + threadIdx.x;
    if (e >= E + n) return;
    int s  = (e < E) ? src[e] : (e - E);
    int dn = (e < E) ? dst[e] : (e - E);
    for (int h = 0; h < heads; ++h) {
        float v = asrc[s * heads + h] + adst[dn * heads + h];
        v = (v > 0.f) ? v : NEGS * v;
        atomicAdd(&zbuf[dn * heads + h], __expf(v - mbuf[dn * heads + h]));
    }
}

// -------------------------------- edge pass 3: weighted scatter (wave/edge)
// One wave32 per edge; lane l handles channels [4l, 4l+4) of the 128-wide
// feature row: float4 gather from h[src], alpha recompute, 4 atomic f32 adds.
__global__ __launch_bounds__(256) void edge_scatter_kernel(
        const int* __restrict__ src, const int* __restrict__ dst,
        const float* __restrict__ asrc, const float* __restrict__ adst,
        const float* __restrict__ mbuf, const float* __restrict__ zbuf,
        const float* __restrict__ hfeat, float* __restrict__ outacc,
        int E, int n, int heads, int d) {
    int wid  = (blockIdx.x * blockDim.x + threadIdx.x) >> 5;
    int lane = threadIdx.x & 31;
    if (wid >= E + n) return;
    int s  = (wid < E) ? src[wid] : (wid - E);
    int dn = (wid < E) ? dst[wid] : (wid - E);
    int c  = lane * 4;              // channel base within 128
    int hd = c / d;                 // head owning these channels
    float v = asrc[s * heads + hd] + adst[dn * heads + hd];
    v = (v > 0.f) ? v : NEGS * v;
    float alpha = __expf(v - mbuf[dn * heads + hd]) /
                  (zbuf[dn * heads + hd] + 1e-16f);
    float4 hv = *(const float4*)(hfeat + (long)s * HC + c);
    float* op = outacc + (long)dn * HC + c;
    atomicAdd(op + 0, hv.x * alpha);
    atomicAdd(op + 1, hv.y * alpha);
    atomicAdd(op + 2, hv.z * alpha);
    atomicAdd(op + 3, hv.w * alpha);
}

// ------------------------------------------------- fused bias + ELU inplace
__global__ void bias_elu_kernel(float* __restrict__ buf,
                                const float* __restrict__ bias, int n) {
    int i = blockIdx.x * blockDim.x + threadIdx.x;
    if (i >= n * HC) return;
    float v = buf[i] + bias[i % HC];
    buf[i] = (v > 0.f) ? v : (__expf(v) - 1.0f);
}

// ------------------------------------------------- final head: out = h@Wlin+b
__global__ void linear_out_kernel(const float* __restrict__ hfeat,
                                  const float* __restrict__ Wlin,
                                  const float* __restrict__ blin,
                                  float* __restrict__ out, int n) {
    int idx = blockIdx.x * blockDim.x + threadIdx.x;
    if (idx >= n * OUTCH) return;
    int node = idx / OUTCH, o = idx % OUTCH;
    const float* hp = hfeat + (long)node * HC;
    float acc = blin[o];
    #pragma unroll 4
    for (int k = 0; k < HC; k += 4) {
        float4 hv = *(const float4*)(hp + k);
        acc += hv.x * Wlin[(k + 0) * OUTCH + o];
        acc += hv.y * Wlin[(k + 1) * OUTCH + o];
        acc += hv.z * Wlin[(k + 2) * OUTCH + o];
        acc += hv.w * Wlin[(k + 3) * OUTCH + o];
    }
    out[idx] = acc;
}

// =============================================================== launcher
extern "C" void kernel_launch(void* const* d_in, const int* in_sizes, int n_in,
                              void* d_out, int out_size, void* d_ws, size_t ws_size,
                              hipStream_t stream) {
    const float* x      = (const float*)d_in[0];
    const int*   ei     = (const int*)  d_in[1];
    const float* W1     = (const float*)d_in[2];
    const float* a_src1 = (const float*)d_in[3];
    const float* a_dst1 = (const float*)d_in[4];
    const float* b1     = (const float*)d_in[5];
    const float* W2     = (const float*)d_in[6];
    const float* a_src2 = (const float*)d_in[7];
    const float* a_dst2 = (const float*)d_in[8];
    const float* b2     = (const float*)d_in[9];
    const float* Wlin   = (const float*)d_in[10];
    const float* blin   = (const float*)d_in[11];
    float* out = (float*)d_out;

    const int n = NNODES;
    const int E = in_sizes[1] / 2;          // edge_index is [2, E]
    const int Et = E + n;                   // with self-loops
    const int* src = ei;
    const int* dst = ei + E;

    // workspace layout (floats)
    float* bufA = (float*)d_ws;             // GEMM output / h features
    float* bufB = bufA + (long)n * HC;      // aggregation accumulator / x2
    float* asrc = bufB + (long)n * HC;
    float* adst = asrc + (long)n * HEADS1;
    float* mbuf = adst + (long)n * HEADS1;
    float* zbuf = mbuf + (long)n * HEADS1;

    const int T = 256;
    auto blocks = [](long work, int t) { return (int)((work + t - 1) / t); };
    const int gemmBlocks = ((n / 16) + 3) / 4;   // 4 waves/block, 1 strip/wave

    // ---------------- layer 1 (heads=4, d=32) ----------------
    fill_kernel<<<blocks((long)n * HC, T), T, 0, stream>>>(bufB, 0.0f, n * HC);
    fill_kernel<<<blocks((long)n * HEADS1, T), T, 0, stream>>>(mbuf, -INFINITY, n * HEADS1);
    fill_kernel<<<blocks((long)n * HEADS1, T), T, 0, stream>>>(zbuf, 0.0f, n * HEADS1);

    gemm128_wmma<<<gemmBlocks, 128, 0, stream>>>(x, W1, bufA, n);
    att_coeff_kernel<<<blocks((long)n * HEADS1, T), T, 0, stream>>>(
        bufA, a_src1, a_dst1, asrc, adst, n, HEADS1, HID1);
    edge_max_kernel<<<blocks(Et, T), T, 0, stream>>>(src, dst, asrc, adst, mbuf, E, n, HEADS1);
    edge_sum_kernel<<<blocks(Et, T), T, 0, stream>>>(src, dst, asrc, adst, mbuf, zbuf, E, n, HEADS1);
    edge_scatter_kernel<<<blocks((long)Et * 32, T), T, 0, stream>>>(
        src, dst, asrc, adst, mbuf, zbuf, bufA, bufB, E, n, HEADS1, HID1);
    bias_elu_kernel<<<blocks((long)n * HC, T), T, 0, stream>>>(bufB, b1, n);

    // ---------------- layer 2 (heads=1, d=128) ----------------
    gemm128_wmma<<<gemmBlocks, 128, 0, stream>>>(bufB, W2, bufA, n);  // reads x2=bufB
    fill_kernel<<<blocks((long)n * HC, T), T, 0, stream>>>(bufB, 0.0f, n * HC);
    fill_kernel<<<blocks((long)n, T), T, 0, stream>>>(mbuf, -INFINITY, n);
    fill_kernel<<<blocks((long)n, T), T, 0, stream>>>(zbuf, 0.0f, n);

    att_coeff_kernel<<<blocks((long)n, T), T, 0, stream>>>(
        bufA, a_src2, a_dst2, asrc, adst, n, 1, HC);
    edge_max_kernel<<<blocks(Et, T), T, 0, stream>>>(src, dst, asrc, adst, mbuf, E, n, 1);
    edge_sum_kernel<<<blocks(Et, T), T, 0, stream>>>(src, dst, asrc, adst, mbuf, zbuf, E, n, 1);
    edge_scatter_kernel<<<blocks((long)Et * 32, T), T, 0, stream>>>(
        src, dst, asrc, adst, mbuf, zbuf, bufA, bufB, E, n, 1, HC);
    bias_elu_kernel<<<blocks((long)n * HC, T), T, 0, stream>>>(bufB, b2, n);

    // ---------------- output head ----------------
    linear_out_kernel<<<blocks((long)n * OUTCH, T), T, 0, stream>>>(bufB, Wlin, blin, out, n);
}